// Plastic_NonModulated_RNN_60163901882586
// MI455X (gfx1250) — compile-verified
//
#include <hip/hip_runtime.h>
#include <hip/hip_bf16.h>

typedef __attribute__((ext_vector_type(2))) float v2f;
typedef __attribute__((ext_vector_type(8))) float v8f;

// ---------------------------------------------------------------------------
// WMMA f32 GEMM:  out[M,N] = A[M,K] @ W[N,K]^T + bias[N]
// One wave computes one 16x16 output tile via V_WMMA_F32_16X16X4_F32,
// stepping K by 4. Exact f32 math (matches reference precision).
// A-matrix 16x4 layout (ISA 7.12.2): lanes 0-15 & 16-31 both hold M=lane%16;
// v0 = K0 (lo half) / K2 (hi half), v1 = K1 / K3. B 4x16 mirrors with
// N = lane%16. C/D: vgpr v -> M = v + 8*(lane/16), N = lane%16.
// ---------------------------------------------------------------------------
__global__ __launch_bounds__(256) void gemm_wmma_f32(
    const float* __restrict__ A,     // [M,K]
    const float* __restrict__ W,     // [N,K]
    const float* __restrict__ bias,  // [N]
    float* __restrict__ out,         // [M,N]
    int M, int K, int N)
{
    int wave = (blockIdx.x * blockDim.x + threadIdx.x) >> 5;
    int lane = threadIdx.x & 31;
    int ntn  = N >> 4;                  // tiles along N
    int bm   = wave / ntn;              // tile row (batch dim)
    int bn   = wave % ntn;              // tile col
    if (bm * 16 >= M) return;           // wave-uniform exit (EXEC stays all-1s in WMMA)

    int half = lane >> 4;               // 0: lanes 0-15, 1: lanes 16-31
    int r    = lane & 15;

    const float* arow = A + (size_t)(bm * 16 + r) * K;
    const float* wrow = W + (size_t)(bn * 16 + r) * K;

    v8f c = {};
    for (int k0 = 0; k0 < K; k0 += 4) {
        float2 af = *(const float2*)(arow + k0 + 2 * half);
        float2 bf = *(const float2*)(wrow + k0 + 2 * half);
        v2f a; a.x = af.x; a.y = af.y;
        v2f b; b.x = bf.x; b.y = bf.y;
        // (neg_a, A, neg_b, B, c_mod, C, reuse_a, reuse_b)
        c = __builtin_amdgcn_wmma_f32_16x16x4_f32(
                false, a, false, b, (short)0, c, false, false);
    }

    float bv = bias[bn * 16 + r];
    #pragma unroll
    for (int v = 0; v < 8; ++v) {
        int mrow = v + 8 * half;
        out[(size_t)(bm * 16 + mrow) * N + (bn * 16 + r)] = c[v] + bv;
    }
}

// ---------------------------------------------------------------------------
// rec[b,i] = sum_j (w[i,j] + alpha[i,j]*hebb[b,i,j]) * prev[b,j]
// hidden[b,i] = tanh(pre[b,i] + rec[b,i])
// One wave32 per (b,i) row; float4 coalesced loads; shuffle reduction.
// This is the pass-1 64 MB hebb read (HBM-bound part #1).
// ---------------------------------------------------------------------------
__global__ __launch_bounds__(256) void rec_hidden_kernel(
    const float* __restrict__ hebb,   // [B,H,H]
    const float* __restrict__ w,      // [H,H]
    const float* __restrict__ alpha,  // [H,H]
    const float* __restrict__ prev,   // [B,H]
    const float* __restrict__ pre,    // [B,H]
    float* __restrict__ hidden)       // [B,H]
{
    const int H = 512;
    int row  = (blockIdx.x * blockDim.x + threadIdx.x) >> 5;  // b*H + i
    int lane = threadIdx.x & 31;
    int b = row >> 9;
    int i = row & 511;

    const float4* hb = (const float4*)(hebb + (size_t)row * H);
    const float4* wr = (const float4*)(w     + (size_t)i   * H);
    const float4* ar = (const float4*)(alpha + (size_t)i   * H);
    const float4* pv = (const float4*)(prev  + (size_t)b   * H);

    float acc = 0.f;
    #pragma unroll
    for (int t = 0; t < 4; ++t) {                 // 128 float4s per row
        int idx = lane + 32 * t;
        float4 hv = hb[idx];
        float4 wv = wr[idx];
        float4 av = ar[idx];
        float4 p  = pv[idx];
        acc += (wv.x + av.x * hv.x) * p.x
             + (wv.y + av.y * hv.y) * p.y
             + (wv.z + av.z * hv.z) * p.z
             + (wv.w + av.w * hv.w) * p.w;
    }
    #pragma unroll
    for (int off = 16; off > 0; off >>= 1)
        acc += __shfl_down(acc, off, 32);
    if (lane == 0)
        hidden[row] = tanhf(pre[row] + acc);
}

// ---------------------------------------------------------------------------
// v_out[b] = hidden[b,:] . Wv[0,:] + bv   (one wave per batch row)
// ---------------------------------------------------------------------------
__global__ __launch_bounds__(256) void vout_kernel(
    const float* __restrict__ hidden,  // [B,H]
    const float* __restrict__ Wv,      // [1,H]
    const float* __restrict__ bv,      // [1]
    float* __restrict__ v_out)         // [B]
{
    const int H = 512;
    int b    = (blockIdx.x * blockDim.x + threadIdx.x) >> 5;
    int lane = threadIdx.x & 31;
    const float4* hr = (const float4*)(hidden + (size_t)b * H);
    const float4* wr = (const float4*)Wv;
    float acc = 0.f;
    #pragma unroll
    for (int t = 0; t < 4; ++t) {
        int idx = lane + 32 * t;
        float4 h = hr[idx];
        float4 wv = wr[idx];
        acc += h.x * wv.x + h.y * wv.y + h.z * wv.z + h.w * wv.w;
    }
    #pragma unroll
    for (int off = 16; off > 0; off >>= 1)
        acc += __shfl_down(acc, off, 32);
    if (lane == 0)
        v_out[b] = acc + bv[0];
}

// ---------------------------------------------------------------------------
// hebb_new = clip(hebb + eta * hidden[b,i] * prev[b,j], -1, 1)
// Pure float4 streaming: 64 MB read (L2-resident from pass 1) + 64 MB write.
// ---------------------------------------------------------------------------
__global__ __launch_bounds__(256) void hebb_update_kernel(
    const float* __restrict__ hebb,    // [B,H,H]
    const float* __restrict__ prev,    // [B,H]
    const float* __restrict__ hidden,  // [B,H]
    const float* __restrict__ eta,     // [1]
    float* __restrict__ hebb_new)      // [B,H,H]
{
    const int H = 512;
    size_t q = (size_t)blockIdx.x * blockDim.x + threadIdx.x;  // float4 index
    size_t e = q * 4;                                          // element index
    int b   = (int)(e >> 18);          // / (512*512)
    int rem = (int)(e & 262143);
    int i   = rem >> 9;
    int j   = rem & 511;

    float s = eta[0] * hidden[b * H + i];
    float4 hv = ((const float4*)hebb)[q];
    float4 p  = ((const float4*)(prev + (size_t)b * H))[j >> 2];
    float4 o;
    o.x = fminf(1.f, fmaxf(-1.f, hv.x + s * p.x));
    o.y = fminf(1.f, fmaxf(-1.f, hv.y + s * p.y));
    o.z = fminf(1.f, fmaxf(-1.f, hv.z + s * p.z));
    o.w = fminf(1.f, fmaxf(-1.f, hv.w + s * p.w));
    ((float4*)hebb_new)[q] = o;
}

extern "C" void kernel_launch(void* const* d_in, const int* in_sizes, int n_in,
                              void* d_out, int out_size, void* d_ws, size_t ws_size,
                              hipStream_t stream) {
    const float* inputs = (const float*)d_in[0];   // [64,256]
    const float* prev   = (const float*)d_in[1];   // [64,512]
    const float* hebb   = (const float*)d_in[2];   // [64,512,512]
    const float* w      = (const float*)d_in[3];   // [512,512]
    const float* alpha  = (const float*)d_in[4];   // [512,512]
    const float* eta    = (const float*)d_in[5];   // [1]
    const float* Wi     = (const float*)d_in[6];   // [512,256]
    const float* bi     = (const float*)d_in[7];   // [512]
    const float* Wo     = (const float*)d_in[8];   // [256,512]
    const float* bo     = (const float*)d_in[9];   // [256]
    const float* Wv     = (const float*)d_in[10];  // [1,512]
    const float* bv     = (const float*)d_in[11];  // [1]

    float* out      = (float*)d_out;
    float* a_out    = out;            // 64*256 = 16384
    float* v_out    = out + 16384;    // 64
    float* hidden   = out + 16448;    // 64*512 = 32768
    float* hebb_new = out + 49216;    // 64*512*512

    float* pre = (float*)d_ws;        // 64*512 floats = 128 KB scratch

    // 1) pre = inputs @ Wi^T + bi   (M=64, K=256, N=512) -> 4*32=128 tiles, 8 waves/block
    gemm_wmma_f32<<<16, 256, 0, stream>>>(inputs, Wi, bi, pre, 64, 256, 512);

    // 2) rec + tanh fused -> hidden  (32768 rows, 1 wave each)
    rec_hidden_kernel<<<4096, 256, 0, stream>>>(hebb, w, alpha, prev, pre, hidden);

    // 3) a_out = hidden @ Wo^T + bo (M=64, K=512, N=256) -> 4*16=64 tiles
    gemm_wmma_f32<<<8, 256, 0, stream>>>(hidden, Wo, bo, a_out, 64, 512, 256);

    // 4) v_out (64 rows, 1 wave each)
    vout_kernel<<<8, 256, 0, stream>>>(hidden, Wv, bv, v_out);

    // 5) hebb_new streaming update: 4,194,304 float4s / 256 threads
    hebb_update_kernel<<<16384, 256, 0, stream>>>(hebb, prev, hidden, eta, hebb_new);
}